// MetaSGDLearningRule_22711787061852
// MI455X (gfx1250) — compile-verified
//
#include <hip/hip_runtime.h>

// Fused per-element Adam (MetaSGD) update for MI455X (gfx1250).
// Memory-bound streaming kernel: ~805 MB traffic/call -> ~35us floor @ 23.3 TB/s.
// Path: coalesced 128-bit non-temporal loads/stores, ILP=2 float4 per thread,
// wave32 blocks of 256. No WMMA (no contraction exists in this op).

typedef __attribute__((ext_vector_type(4))) float v4f;

#define ADAM_BETA1 0.9f
#define ADAM_BETA2 0.99f
#define ADAM_OMB1 0.1f    // 1 - beta1 (as the f32 the reference traces)
#define ADAM_OMB2 0.01f   // 1 - beta2
#define ADAM_EPS 1e-8f

__device__ __forceinline__ v4f adam_vec(v4f W, v4f G, v4f LR, v4f M, v4f V,
                                        float inv_bc1, float inv_sq_bc2) {
  v4f O;
#pragma unroll
  for (int c = 0; c < 4; ++c) {
    float mn = ADAM_BETA1 * M[c] + ADAM_OMB1 * G[c];
    float vn = ADAM_BETA2 * V[c] + ADAM_OMB2 * (G[c] * G[c]);
    float denom = sqrtf(vn) * inv_sq_bc2 + ADAM_EPS;
    O[c] = W[c] - (LR[c] * inv_bc1) * mn / denom;
  }
  return O;
}

__global__ __launch_bounds__(256) void adam_update_vec4(
    const float* __restrict__ w, const float* __restrict__ g,
    const float* __restrict__ lr, const float* __restrict__ m,
    const float* __restrict__ v, float* __restrict__ out,
    int n4, const int* __restrict__ step_ptr) {
  // Uniform scalar: step lives in device memory (graph capture forbids host read).
  const int step = *step_ptr;
  float b1p = 1.0f, b2p = 1.0f;
  for (int s = 0; s < step; ++s) { b1p *= ADAM_BETA1; b2p *= ADAM_BETA2; }
  const float inv_bc1    = 1.0f / (1.0f - b1p);
  const float inv_sq_bc2 = 1.0f / sqrtf(1.0f - b2p);

  const v4f* __restrict__ w4  = (const v4f*)w;
  const v4f* __restrict__ g4  = (const v4f*)g;
  const v4f* __restrict__ lr4 = (const v4f*)lr;
  const v4f* __restrict__ m4  = (const v4f*)m;
  const v4f* __restrict__ v4  = (const v4f*)v;
  v4f* __restrict__ o4        = (v4f*)out;

  const int tile = blockDim.x * 2;                 // 2 float4 chunks per thread
  const int base = blockIdx.x * tile;
  const int i0 = base + threadIdx.x;
  const int i1 = i0 + blockDim.x;

  if (base + tile <= n4) {
    // Fast path: full tile, no guards -> loads cluster into 10 b128 in flight.
    v4f W0 = __builtin_nontemporal_load(w4  + i0);
    v4f G0 = __builtin_nontemporal_load(g4  + i0);
    v4f L0 = __builtin_nontemporal_load(lr4 + i0);
    v4f M0 = __builtin_nontemporal_load(m4  + i0);
    v4f V0 = __builtin_nontemporal_load(v4  + i0);
    v4f W1 = __builtin_nontemporal_load(w4  + i1);
    v4f G1 = __builtin_nontemporal_load(g4  + i1);
    v4f L1 = __builtin_nontemporal_load(lr4 + i1);
    v4f M1 = __builtin_nontemporal_load(m4  + i1);
    v4f V1 = __builtin_nontemporal_load(v4  + i1);
    v4f O0 = adam_vec(W0, G0, L0, M0, V0, inv_bc1, inv_sq_bc2);
    v4f O1 = adam_vec(W1, G1, L1, M1, V1, inv_bc1, inv_sq_bc2);
    __builtin_nontemporal_store(O0, o4 + i0);
    __builtin_nontemporal_store(O1, o4 + i1);
  } else {
    // Tail tile (never taken for the reference shapes, all multiples of 512 vec4s
    // except the final 2048-vec4 tensor which still tiles exactly; kept for safety).
    if (i0 < n4) {
      v4f O = adam_vec(__builtin_nontemporal_load(w4 + i0),
                       __builtin_nontemporal_load(g4 + i0),
                       __builtin_nontemporal_load(lr4 + i0),
                       __builtin_nontemporal_load(m4 + i0),
                       __builtin_nontemporal_load(v4 + i0),
                       inv_bc1, inv_sq_bc2);
      __builtin_nontemporal_store(O, o4 + i0);
    }
    if (i1 < n4) {
      v4f O = adam_vec(__builtin_nontemporal_load(w4 + i1),
                       __builtin_nontemporal_load(g4 + i1),
                       __builtin_nontemporal_load(lr4 + i1),
                       __builtin_nontemporal_load(m4 + i1),
                       __builtin_nontemporal_load(v4 + i1),
                       inv_bc1, inv_sq_bc2);
      __builtin_nontemporal_store(O, o4 + i1);
    }
  }
}

// Scalar tail kernel for element counts not divisible by 4 (not hit with the
// reference shapes; included so the launcher is shape-generic).
__global__ __launch_bounds__(256) void adam_update_scalar(
    const float* __restrict__ w, const float* __restrict__ g,
    const float* __restrict__ lr, const float* __restrict__ m,
    const float* __restrict__ v, float* __restrict__ out,
    int start, int n, const int* __restrict__ step_ptr) {
  int i = start + blockIdx.x * blockDim.x + threadIdx.x;
  if (i >= n) return;
  const int step = *step_ptr;
  float b1p = 1.0f, b2p = 1.0f;
  for (int s = 0; s < step; ++s) { b1p *= ADAM_BETA1; b2p *= ADAM_BETA2; }
  const float inv_bc1    = 1.0f / (1.0f - b1p);
  const float inv_sq_bc2 = 1.0f / sqrtf(1.0f - b2p);
  float mn = ADAM_BETA1 * m[i] + ADAM_OMB1 * g[i];
  float vn = ADAM_BETA2 * v[i] + ADAM_OMB2 * (g[i] * g[i]);
  float denom = sqrtf(vn) * inv_sq_bc2 + ADAM_EPS;
  out[i] = w[i] - (lr[i] * inv_bc1) * mn / denom;
}

extern "C" void kernel_launch(void* const* d_in, const int* in_sizes, int n_in,
                              void* d_out, int out_size, void* d_ws, size_t ws_size,
                              hipStream_t stream) {
  (void)n_in; (void)out_size; (void)d_ws; (void)ws_size;
  const int* step_ptr = (const int*)d_in[15];
  float* out = (float*)d_out;

  size_t offset = 0;
  for (int p = 0; p < 3; ++p) {
    const float* w  = (const float*)d_in[5 * p + 0];
    const float* g  = (const float*)d_in[5 * p + 1];
    const float* lr = (const float*)d_in[5 * p + 2];
    const float* m  = (const float*)d_in[5 * p + 3];
    const float* v  = (const float*)d_in[5 * p + 4];
    const int n  = in_sizes[5 * p + 0];
    const int n4 = n >> 2;

    if (n4 > 0) {
      const int block = 256;
      const int tile = block * 2;               // 2 float4 per thread
      const int grid = (n4 + tile - 1) / tile;
      adam_update_vec4<<<grid, block, 0, stream>>>(
          w, g, lr, m, v, out + offset, n4, step_ptr);
    }
    const int rem_start = n4 << 2;
    if (rem_start < n) {
      adam_update_scalar<<<1, 256, 0, stream>>>(
          w, g, lr, m, v, out + offset, rem_start, n, step_ptr);
    }
    offset += (size_t)n;
  }
}